// TaylorMap_12232066859102
// MI455X (gfx1250) — compile-verified
//
#include <hip/hip_runtime.h>
#include <cstdint>
#include <cstddef>

// ---------------------------------------------------------------------------
// TaylorMap: out = poly3(X) @ W via factored monomials.
//   deg3 monomial = x_j * P2[q]  (P2 = degree-2 block, prefix structure)
// Kernel 1 (repack): W -> padded, zero-filled, K-pair-interleaved copy in ws.
// Kernel 2 (main): 64 rows/wave (4 M-tiles); P2 staged in LDS in 6 phases.
// Per K-chunk step: 4x ds_load_b64 (P2 pairs) * x_j + 2x global_load_b64 (B)
// feeding 8x v_wmma_f32_16x16x4_f32.
// ---------------------------------------------------------------------------

#define NF      32
#define NPOLY   6545
#define PITCHX  34            // x row pitch (even; 17L mod 32 distinct -> conflict-free)
#define P2PITCH 90            // 88-float window + pad; 26L mod 64 distinct -> conflict-free
#define P2WIN   88            // floats of P2 per phase
#define NPHASE  6             // 6*88 = 528 = full degree-2 block
#define CHPP    22            // K-chunks (of 4) per phase
#define WAVES   4
#define TPB     (WAVES * 32)
#define ROWS_PER_WAVE  64
#define ROWS_PER_BLOCK (WAVES * ROWS_PER_WAVE)   // 256
#define WAVE_LDS_FLTS  (ROWS_PER_WAVE * PITCHX + ROWS_PER_WAVE * P2PITCH)  // 7936
#define LDS_BYTES      (WAVES * WAVE_LDS_FLTS * 4)                          // 126976

typedef float v2f __attribute__((ext_vector_type(2)));
typedef float v8f __attribute__((ext_vector_type(8)));

// ---------------------------------------------------------------------------
// Compile-time tables
// ---------------------------------------------------------------------------
struct Tables {
  int      doff[33];     // padded dst row offset of deg3 block j; doff[32] = total rows
  int      ch3[32];      // roundup4(c_j)/4 : K-chunk count of deg3 block j
  uint32_t pair[528];    // degree-2 monomial (i | j2<<8) in reference order
  int      srcrow[6596]; // padded row -> original W row (-1 => zero row)
};

static constexpr Tables make_tables() {
  Tables t{};
  int q = 0;
  for (int j2 = 0; j2 < NF; ++j2)
    for (int i = 0; i <= j2; ++i)
      t.pair[q++] = (uint32_t)i | ((uint32_t)j2 << 8);

  int off = 564;  // 4 (deg0 padded) + 32 (deg1) + 528 (deg2)
  for (int j = 0; j < NF; ++j) {
    const int c  = (j + 1) * (j + 2) / 2;
    const int r4 = (c + 3) & ~3;
    t.doff[j] = off;
    t.ch3[j]  = r4 / 4;
    off += r4;
  }
  t.doff[32] = off;  // 6596

  int r = 0;
  t.srcrow[r++] = 0;                       // deg0
  t.srcrow[r++] = -1; t.srcrow[r++] = -1; t.srcrow[r++] = -1;
  for (int j = 0; j < NF; ++j) t.srcrow[r++] = 1 + j;        // deg1
  for (q = 0; q < 528; ++q)    t.srcrow[r++] = 33 + q;       // deg2
  int s = 561;                                               // deg3
  for (int j = 0; j < NF; ++j) {
    const int c  = (j + 1) * (j + 2) / 2;
    const int r4 = (c + 3) & ~3;
    for (q = 0; q < r4; ++q) t.srcrow[r++] = (q < c) ? (s + q) : -1;
    s += c;
  }
  return t;
}

static constexpr Tables HTAB = make_tables();
static constexpr int NPADROWS = 6596;   // == HTAB.doff[32]

__constant__ Tables TAB = make_tables();

// ---------------------------------------------------------------------------
// Kernel 1: repack W into padded + pair-interleaved layout in workspace.
//   WI[(p>>1)*64 + f*2 + (p&1)] = Wpadded[p, f]
// so a b64 load at (pairIdx*64 + f*2) yields (W[p,f], W[p+1,f]).
// ---------------------------------------------------------------------------
__global__ __launch_bounds__(256) void repack_w(const float* __restrict__ W,
                                                float* __restrict__ WI,
                                                int total) {
  const int t = blockIdx.x * 256 + threadIdx.x;
  if (t >= total) return;
  const int r = t >> 5;
  const int f = t & 31;
  const int s = TAB.srcrow[r];
  const float v = (s >= 0) ? W[(size_t)s * NF + f] : 0.0f;
  WI[(size_t)(r >> 1) * 64 + f * 2 + (r & 1)] = v;
}

// ---------------------------------------------------------------------------
// Kernel 2: main GEMM with on-the-fly factored A generation.
// ---------------------------------------------------------------------------
#define WMMA1(C, A, B)                                                        \
  C = __builtin_amdgcn_wmma_f32_16x16x4_f32(false, (A), false, (B),           \
                                            (short)0, C, false, false)

#define WMMA8(A0, A1, A2, A3)                                                 \
  WMMA1(cA0, A0, b0); WMMA1(cA1, A0, b1);                                     \
  WMMA1(cB0, A1, b0); WMMA1(cB1, A1, b1);                                     \
  WMMA1(cC0, A2, b0); WMMA1(cC1, A2, b1);                                     \
  WMMA1(cD0, A3, b0); WMMA1(cD1, A3, b1)

__global__ __launch_bounds__(TPB) void
taylor_wmma_f32(const float* __restrict__ X,
                const float* __restrict__ WI,
                float* __restrict__ out) {
  extern __shared__ float smem[];

  const int tid  = threadIdx.x;
  const int wave = tid >> 5;
  const int lane = tid & 31;
  const int hi   = lane >> 4;    // selects K pair {2hi, 2hi+1} within a chunk
  const int m    = lane & 15;    // A row / B,C column within tile

  const int rowBase = blockIdx.x * ROWS_PER_BLOCK + wave * ROWS_PER_WAVE;
  float* xw  = smem + wave * WAVE_LDS_FLTS;
  float* p2w = xw + ROWS_PER_WAVE * PITCHX;

  // Stage this wave's 64 rows of X into LDS (coalesced).
  for (int i = lane; i < ROWS_PER_WAVE * NF; i += 32) {
    const int r = i >> 5, c = i & 31;
    xw[r * PITCHX + c] = X[(size_t)(rowBase + r) * NF + c];
  }

  // Per-M-tile row pointers (rows m, m+16, m+32, m+48).
  const float* xrow0 = &xw[m * PITCHX];
  const float* xrow1 = xrow0 + 16 * PITCHX;
  const float* xrow2 = xrow0 + 32 * PITCHX;
  const float* xrow3 = xrow0 + 48 * PITCHX;
  // K-pair-adjusted pointers (include the 2*hi element offset).
  const float* xq0 = xrow0 + 2 * hi;
  const float* xq1 = xrow1 + 2 * hi;
  const float* xq2 = xrow2 + 2 * hi;
  const float* xq3 = xrow3 + 2 * hi;
  const float* pq0 = &p2w[m * P2PITCH] + 2 * hi;
  const float* pq1 = pq0 + 16 * P2PITCH;
  const float* pq2 = pq0 + 32 * P2PITCH;
  const float* pq3 = pq0 + 48 * P2PITCH;
  // Generation pointers: lane builds rows (lane) and (lane+32).
  const float* gx0 = &xw[lane * PITCHX];
  const float* gx1 = gx0 + 32 * PITCHX;
  float* gp0 = &p2w[lane * P2PITCH];
  float* gp1 = gp0 + 32 * P2PITCH;

  v8f cA0 = {}, cA1 = {}, cB0 = {}, cB1 = {};
  v8f cC0 = {}, cC1 = {}, cD0 = {}, cD1 = {};
  v2f a0, a1, a2, a3, b0, b1;

  // B fetch: pairIdx -> (Wp[p,f], Wp[p+1,f]) per lane, both N-tiles.
  #define LOADB(PAIRIDX)                                                   \
    do {                                                                   \
      const float* _wb = WI + (size_t)(PAIRIDX) * 64 + m * 2;              \
      b0 = *(const v2f*)(_wb);                                             \
      b1 = *(const v2f*)(_wb + 32);                                        \
    } while (0)

  // ---- degree 0 (padded rows 0..3): A = [1,0,0,0] along K ----
  a0.x = hi ? 0.0f : 1.0f;  a0.y = 0.0f;
  LOADB(hi);
  WMMA8(a0, a0, a0, a0);

  // ---- degree 1 (padded rows 4..35): A pairs straight from X rows ----
#pragma unroll
  for (int c2 = 0; c2 < 8; ++c2) {
    a0 = *(const v2f*)(xq0 + 4 * c2);
    a1 = *(const v2f*)(xq1 + 4 * c2);
    a2 = *(const v2f*)(xq2 + 4 * c2);
    a3 = *(const v2f*)(xq3 + 4 * c2);
    LOADB(2 + 2 * c2 + hi);
    WMMA8(a0, a1, a2, a3);
  }

  // ---- degrees 2 and 3, phased over the P2 window ----
  for (int ph = 0; ph < NPHASE; ++ph) {
    // Generate P2 window q in [88*ph, 88*ph+88): lane builds rows lane, lane+32.
    const int qbase = P2WIN * ph;
    for (int qq = 0; qq < P2WIN; ++qq) {
      const uint32_t e = TAB.pair[qbase + qq];               // uniform -> s_load
      const int fi = e & 255, fj = (e >> 8) & 255;
      gp0[qq] = gx0[fi] * gx0[fj];
      gp1[qq] = gx1[fi] * gx1[fj];
    }
    // (wave-private region; LDS ops are in-order per wave -> no barrier)

    // degree 2: out += P2 @ W2  (chunks 22*ph .. 22*ph+21)
#pragma unroll 2
    for (int lc = 0; lc < CHPP; ++lc) {
      a0 = *(const v2f*)(pq0 + 4 * lc);
      a1 = *(const v2f*)(pq1 + 4 * lc);
      a2 = *(const v2f*)(pq2 + 4 * lc);
      a3 = *(const v2f*)(pq3 + 4 * lc);
      LOADB(18 + 2 * (CHPP * ph + lc) + hi);
      WMMA8(a0, a1, a2, a3);
    }

    // degree 3: for each j, out += diag(x_j) * P2_prefix @ W3_j
    for (int j = 0; j < NF; ++j) {
      int nch = TAB.ch3[j] - CHPP * ph;        // uniform
      if (nch <= 0) continue;                   // uniform branch, EXEC intact
      if (nch > CHPP) nch = CHPP;
      const float xs0 = xrow0[j], xs1 = xrow1[j];
      const float xs2 = xrow2[j], xs3 = xrow3[j];
      const v2f xj0 = {xs0, xs0}, xj1 = {xs1, xs1};
      const v2f xj2 = {xs2, xs2}, xj3 = {xs3, xs3};
      const int pbase = (TAB.doff[j] >> 1) + 2 * (CHPP * ph) + hi;
#pragma unroll 2
      for (int lc = 0; lc < nch; ++lc) {
        a0 = *(const v2f*)(pq0 + 4 * lc) * xj0;   // v_pk_mul_f32
        a1 = *(const v2f*)(pq1 + 4 * lc) * xj1;
        a2 = *(const v2f*)(pq2 + 4 * lc) * xj2;
        a3 = *(const v2f*)(pq3 + 4 * lc) * xj3;
        LOADB(pbase + 2 * lc);
        WMMA8(a0, a1, a2, a3);
      }
    }
  }

  // ---- store: VGPR r, lanes 0-15 -> M=r, lanes 16-31 -> M=r+8; N = m ----
#pragma unroll
  for (int r = 0; r < 8; ++r) {
    const int row0 = rowBase + r + 8 * hi;
    out[(size_t)row0 * NF + m]             = cA0[r];
    out[(size_t)row0 * NF + 16 + m]        = cA1[r];
    out[(size_t)(row0 + 16) * NF + m]      = cB0[r];
    out[(size_t)(row0 + 16) * NF + 16 + m] = cB1[r];
    out[(size_t)(row0 + 32) * NF + m]      = cC0[r];
    out[(size_t)(row0 + 32) * NF + 16 + m] = cC1[r];
    out[(size_t)(row0 + 48) * NF + m]      = cD0[r];
    out[(size_t)(row0 + 48) * NF + 16 + m] = cD1[r];
  }
  #undef LOADB
}

// ---------------------------------------------------------------------------
extern "C" void kernel_launch(void* const* d_in, const int* in_sizes, int n_in,
                              void* d_out, int out_size, void* d_ws, size_t ws_size,
                              hipStream_t stream) {
  const float* X = (const float*)d_in[0];   // (BATCH, 32)
  const float* W = (const float*)d_in[1];   // (6545, 32)
  float* out = (float*)d_out;               // (BATCH, 32)
  float* WI  = (float*)d_ws;                // 6596*32 floats = 844 KB

  // opt-in to >48KB dynamic LDS (idempotent, capture-safe host call)
  hipFuncSetAttribute((const void*)taylor_wmma_f32,
                      hipFuncAttributeMaxDynamicSharedMemorySize, LDS_BYTES);

  const int totalRepack = NPADROWS * NF;                    // 211072
  repack_w<<<(totalRepack + 255) / 256, 256, 0, stream>>>(W, WI, totalRepack);

  const int batch  = in_sizes[0] / NF;                      // 32768
  const int blocks = batch / ROWS_PER_BLOCK;                // 128
  taylor_wmma_f32<<<blocks, TPB, LDS_BYTES, stream>>>(X, WI, out);
}